// LSTM_DIST_45878840656276
// MI455X (gfx1250) — compile-verified
//
#include <hip/hip_runtime.h>

// Problem constants (match reference):
//   lstm_out: [B=32, T=2048, H2=512] f32, fwd = [:, :, :256], bwd = [:, :, 256:]
//   spans:    [B, S=128, 2] i32 (two sets: am, ac)
//   out:      [B, S, 1536] f32 = concat(am_rep[768], ac_rep[768])
namespace {
constexpr int Bc  = 32;
constexpr int Tc  = 2048;
constexpr int H2c = 512;
constexpr int Hc  = 256;    // half hidden
constexpr int Sc  = 128;
constexpr int REP = 3 * Hc;      // 768 floats per span representation
constexpr int OUTW = 2 * REP;    // 1536 floats per (b, s) output row
}

__device__ __forceinline__ float4 f4sub(float4 a, float4 b) {
    return make_float4(a.x - b.x, a.y - b.y, a.z - b.z, a.w - b.w);
}

// One 64-thread block (2 wave32) per (set, b, s) span.
// Thread t handles floats [4t, 4t+4) of the H=256 hidden slice:
//   4x global_load_b128 (gathered rows), 3x global_store_b128 (contiguous out).
__global__ __launch_bounds__(64)
void lstm_minus_span_kernel(const float* __restrict__ lstm,
                            const int*   __restrict__ am_spans,
                            const int*   __restrict__ ac_spans,
                            float*       __restrict__ out)
{
    const int q   = blockIdx.x;        // [0, 2*B*S)
    const int set = q >> 12;           // B*S = 4096: 0 = am, 1 = ac
    const int r   = q & 4095;
    const int b   = r >> 7;            // S = 128
    const int s   = r & 127;

    const int* __restrict__ spans = (set == 0) ? am_spans : ac_spans;
    // Block-uniform span endpoints -> scalar loads, uniform branches.
    const int i = spans[((b * Sc) + s) * 2 + 0];
    const int j = spans[((b * Sc) + s) * 2 + 1];

    const int    t4      = threadIdx.x << 2;     // float offset within H slice
    const size_t rowbase = (size_t)b * Tc;       // row index base for batch b

    // f_end = fwd[b, j], b_start = bwd[b, i]  (always valid)
    const float4 f_end   = *(const float4*)(lstm + (rowbase + (size_t)j) * H2c + t4);
    const float4 b_start = *(const float4*)(lstm + (rowbase + (size_t)i) * H2c + Hc + t4);

    // f_pre = fwd[b, i-1] if i > 0 else 0
    float4 f_pre = make_float4(0.f, 0.f, 0.f, 0.f);
    if (i > 0)
        f_pre = *(const float4*)(lstm + (rowbase + (size_t)(i - 1)) * H2c + t4);

    // b_post = bwd[b, j+1] if j+1 < T else 0
    float4 b_post = make_float4(0.f, 0.f, 0.f, 0.f);
    if (j + 1 < Tc)
        b_post = *(const float4*)(lstm + (rowbase + (size_t)(j + 1)) * H2c + Hc + t4);

    float4 r0 = f4sub(f_end,   f_pre);    // f_j - f_{i-1}
    float4 r1 = f4sub(b_start, b_post);   // b_i - b_{j+1}
    float4 r2 = f4sub(f_pre,   b_post);   // f_{i-1} - b_{j+1}

    // (i, j) == (0, 0) padding spans -> all-zero row
    if ((i == 0) && (j == 0)) {
        r0 = make_float4(0.f, 0.f, 0.f, 0.f);
        r1 = r0;
        r2 = r0;
    }

    float* __restrict__ o =
        out + ((size_t)b * Sc + s) * OUTW + (size_t)set * REP + t4;
    *(float4*)(o)           = r0;   // [0,   256)
    *(float4*)(o + Hc)      = r1;   // [256, 512)
    *(float4*)(o + 2 * Hc)  = r2;   // [512, 768)
}

extern "C" void kernel_launch(void* const* d_in, const int* in_sizes, int n_in,
                              void* d_out, int out_size, void* d_ws, size_t ws_size,
                              hipStream_t stream) {
    (void)in_sizes; (void)n_in; (void)out_size; (void)d_ws; (void)ws_size;

    const float* lstm = (const float*)d_in[0];   // [B, T, 512] f32
    const int*   am   = (const int*)d_in[1];     // [B, S, 2] i32
    const int*   ac   = (const int*)d_in[2];     // [B, S, 2] i32
    float*       out  = (float*)d_out;           // [B, S, 1536] f32

    const int nblocks = 2 * Bc * Sc;             // 8192 spans
    lstm_minus_span_kernel<<<nblocks, 64, 0, stream>>>(lstm, am, ac, out);
}